// LSTM_1030792151143
// MI455X (gfx1250) — compile-verified
//
#include <hip/hip_runtime.h>

// ---------------------------------------------------------------------------
// Fused 2-layer LSTM + tanh + FC + sigmoid for MI455X (gfx1250, wave32).
//
// One wave owns a 16-row batch tile and walks all T=1000 steps. Per step:
//   - 20 V_WMMA_F32_16X16X4_F32 compute xg(t+1) (input proj, K=40) from
//     X loads issued at the top of the iteration (software pipelined, so
//     global latency hides under the step-t serial chain),
//   - 4 WMMAs for the layer-1 recurrence (seeded with precomputed xg(t)),
//   - 8 WMMAs for layer-2 (gates padded 24->32 cols, K padded 6->8 with
//     zero weights so pad lanes stay exactly 0).
// Serial chain per step is just: ds_load h -> 2 WMMA -> branchless
// activations (v_exp + v_rcp) -> ds_store. State transposes go through
// padded LDS tiles written by all 32 lanes (no EXEC divergence).
// ---------------------------------------------------------------------------

#define B_  2048
#define T_  1000
#define F_  40
#define H1_ 8
#define H2_ 6

typedef __attribute__((ext_vector_type(2))) float v2f;
typedef __attribute__((ext_vector_type(8))) float v8f;

__device__ __forceinline__ v8f wmma_k4(v2f a, v2f b, v8f c) {
  return __builtin_amdgcn_wmma_f32_16x16x4_f32(false, a, false, b, (short)0, c,
                                               false, false);
}

// sigmoid via v_exp_f32 + v_rcp_f32 (no precise-div expansion)
__device__ __forceinline__ float fastsig(float x) {
  return __builtin_amdgcn_rcpf(1.0f + __expf(-x));
}
// tanh(x) = 1 - 2/(exp(2x)+1); saturates to +-1, no NaN paths.
__device__ __forceinline__ float fasttanh(float x) {
  return 1.0f - 2.0f * __builtin_amdgcn_rcpf(__expf(2.0f * x) + 1.0f);
}

__global__ __launch_bounds__(32) void lstm_fused_kernel(
    const float* __restrict__ x,
    const float* __restrict__ Wih1, const float* __restrict__ Whh1,
    const float* __restrict__ bih1, const float* __restrict__ bhh1,
    const float* __restrict__ Wih2, const float* __restrict__ Whh2,
    const float* __restrict__ bih2, const float* __restrict__ bhh2,
    const float* __restrict__ Wfc,  const float* __restrict__ bfc,
    float* __restrict__ out) {
  const int lane = threadIdx.x;   // 0..31
  const int n16  = lane & 15;     // N column within a 16-wide tile / batch row
  const int hi   = lane >> 4;     // half select (K pair / M+8)
  const int b0   = blockIdx.x * 16;

  // Padded to 16 columns: cols 0..7 are real state, cols 8..15 are
  // write-only scratch so ALL lanes can store without EXEC divergence.
  __shared__ __align__(16) float h1s[16][16];
  __shared__ __align__(16) float a1s[16][16];
  __shared__ __align__(16) float h2s[16][16];

  for (int i = lane; i < 16 * 16; i += 32) {
    (&h1s[0][0])[i] = 0.0f;
    (&a1s[0][0])[i] = 0.0f;
    (&h2s[0][0])[i] = 0.0f;
  }
  __syncthreads();

  // ---- preload WMMA B-operands (weights, time-invariant) ----
  v2f  Bx1[2][10], Bh1[2][2];
  float bias1[2];
#pragma unroll
  for (int tt = 0; tt < 2; ++tt) {
    const int n = 16 * tt + n16;
#pragma unroll
    for (int c = 0; c < 10; ++c)
      Bx1[tt][c] = *(const v2f*)(Wih1 + n * F_ + 4 * c + 2 * hi);
#pragma unroll
    for (int c = 0; c < 2; ++c)
      Bh1[tt][c] = *(const v2f*)(Whh1 + n * H1_ + 4 * c + 2 * hi);
    bias1[tt] = bih1[n] + bhh1[n];
  }

  // Layer-2: gate columns padded 24->32 (stride-8 groups i|f|g|o), K padded
  // 6->8. Pad weights/biases are zero -> pad state stays exactly zero.
  v2f  Bx2[2][2], Bh2[2][2];
  float bias2[2];
  const int jj = n16 & 7;
#pragma unroll
  for (int tt = 0; tt < 2; ++tt) {
    const int  grp   = 2 * tt + (n16 >> 3);  // 0..3 = i,f,g,o
    const int  srow  = grp * H2_ + jj;
    const bool valid = (jj < H2_);
#pragma unroll
    for (int c = 0; c < 2; ++c) {
      const int k0 = 4 * c + 2 * hi;
      v2f w;
      w.x = valid ? Wih2[srow * H1_ + k0]     : 0.0f;
      w.y = valid ? Wih2[srow * H1_ + k0 + 1] : 0.0f;
      Bx2[tt][c] = w;
      v2f w2;
      w2.x = (valid && k0     < H2_) ? Whh2[srow * H2_ + k0]     : 0.0f;
      w2.y = (valid && k0 + 1 < H2_) ? Whh2[srow * H2_ + k0 + 1] : 0.0f;
      Bh2[tt][c] = w2;
    }
    bias2[tt] = valid ? (bih2[srow] + bhh2[srow]) : 0.0f;
  }

  const float wfc = (n16 < H2_) ? Wfc[n16] : 0.0f;
  const float bf  = bfc[0];

  // Branchless gate nonlinearity for the g|o tile:
  //   lanes n16<8  (g gate, tanh):    2*sig(2x) - 1
  //   lanes n16>=8 (o gate, sigmoid): 1*sig(1x) - 0
  const float tsc  = (n16 < 8) ? 2.0f : 1.0f;
  const float tsub = (n16 < 8) ? 1.0f : 0.0f;

  float c1[8], c2[8], h2r[8];
#pragma unroll
  for (int r = 0; r < 8; ++r) { c1[r] = 0.0f; c2[r] = 0.0f; h2r[r] = 0.0f; }

  // A-operand pointer for x: lane holds X[b0+n16][t][4c + 2*hi + {0,1}]
  const float* xrow = x + (size_t)(b0 + n16) * (T_ * F_) + 2 * hi;

  // ---- prologue: xg(0) = bias + X_0 @ W_ih1^T ----
  v8f xg0, xg1;
#pragma unroll
  for (int r = 0; r < 8; ++r) { xg0[r] = bias1[0]; xg1[r] = bias1[1]; }
  {
    v2f ax0[10];
#pragma unroll
    for (int c = 0; c < 10; ++c) ax0[c] = *(const v2f*)(xrow + 4 * c);
#pragma unroll
    for (int c = 0; c < 10; ++c) {
      xg0 = wmma_k4(ax0[c], Bx1[0][c], xg0);
      xg1 = wmma_k4(ax0[c], Bx1[1][c], xg1);
    }
  }

  for (int t = 0; t < T_; ++t) {
    // ---- issue next-step X loads immediately (consumed late in body) ----
    const float* xnext = (t + 1 < T_) ? (xrow + F_) : xrow;  // clamp OOB
    __builtin_prefetch(xnext + 8 * F_, 0, 0);
    v2f axn[10];
#pragma unroll
    for (int c = 0; c < 10; ++c) axn[c] = *(const v2f*)(xnext + 4 * c);

    // ---- A-operands from previous step's state (ds_load_2addr_b64) ----
    v2f ah[2], ar2[2];
#pragma unroll
    for (int c = 0; c < 2; ++c) {
      ah[c]  = *(const v2f*)(&h1s[n16][4 * c + 2 * hi]);
      ar2[c] = *(const v2f*)(&h2s[n16][4 * c + 2 * hi]);
    }

    // ---- layer-2 recurrence WMMAs (independent of layer-1 chain) ----
    v8f d0, d1;
#pragma unroll
    for (int r = 0; r < 8; ++r) { d0[r] = bias2[0]; d1[r] = bias2[1]; }
#pragma unroll
    for (int c = 0; c < 2; ++c) {
      d0 = wmma_k4(ar2[c], Bh2[0][c], d0);
      d1 = wmma_k4(ar2[c], Bh2[1][c], d1);
    }

    // ---- layer-1 gates: precomputed xg(t) + H recurrence ----
    v8f acc0 = xg0, acc1 = xg1;
#pragma unroll
    for (int c = 0; c < 2; ++c) {
      acc0 = wmma_k4(ah[c], Bh1[0][c], acc0);
      acc1 = wmma_k4(ah[c], Bh1[1][c], acc1);
    }

    // tile0 cols = i|f (sigmoid), tile1 = g|o (tanh|sigmoid, branchless).
    float h1r[8];
#pragma unroll
    for (int r = 0; r < 8; ++r) {
      const float s0 = fastsig(acc0[r]);                    // i or f
      const float s1 = tsc * fastsig(tsc * acc1[r]) - tsub; // g or o
      const float fo = __shfl(s0, lane ^ 8, 32);            // partner f
      const float oo = __shfl(s1, lane ^ 8, 32);            // partner o
      const float cn = fo * c1[r] + s0 * s1;                // valid in n16<8
      c1[r]  = cn;
      h1r[r] = oo * fasttanh(cn);
    }
    // Unconditional stores (cols 8..15 are scratch for lanes n16>=8).
#pragma unroll
    for (int r = 0; r < 8; ++r) {
      const float h = h1r[r];
      h1s[r + 8 * hi][n16] = h;
      a1s[r + 8 * hi][n16] = fasttanh(h);
    }
    __syncthreads();

    // ---- xg(t+1): independent of everything below; co-executes ----
#pragma unroll
    for (int r = 0; r < 8; ++r) { xg0[r] = bias1[0]; xg1[r] = bias1[1]; }
#pragma unroll
    for (int c = 0; c < 10; ++c) {
      xg0 = wmma_k4(axn[c], Bx1[0][c], xg0);
      xg1 = wmma_k4(axn[c], Bx1[1][c], xg1);
    }

    // ---- layer-2 input projection from a1(t) ----
#pragma unroll
    for (int c = 0; c < 2; ++c) {
      const v2f aa = *(const v2f*)(&a1s[n16][4 * c + 2 * hi]);
      d0 = wmma_k4(aa, Bx2[0][c], d0);
      d1 = wmma_k4(aa, Bx2[1][c], d1);
    }

#pragma unroll
    for (int r = 0; r < 8; ++r) {
      const float s0 = fastsig(d0[r]);
      const float s1 = tsc * fastsig(tsc * d1[r]) - tsub;
      const float fo = __shfl(s0, lane ^ 8, 32);
      const float oo = __shfl(s1, lane ^ 8, 32);
      const float cn = fo * c2[r] + s0 * s1;
      c2[r]  = cn;
      h2r[r] = oo * fasttanh(cn);
    }
#pragma unroll
    for (int r = 0; r < 8; ++r) h2s[r + 8 * hi][n16] = h2r[r];
    __syncthreads();

    xrow = xnext;
  }

  // ---------------- FC + sigmoid on tanh(h2_{T-1}) ----------------
  float p[8];
#pragma unroll
  for (int r = 0; r < 8; ++r)
    p[r] = (n16 < H2_) ? fasttanh(h2r[r]) * wfc : 0.0f;
#pragma unroll
  for (int m = 1; m <= 4; m <<= 1) {
#pragma unroll
    for (int r = 0; r < 8; ++r) p[r] += __shfl_xor(p[r], m, 32);
  }
  if (n16 == 0) {
#pragma unroll
    for (int r = 0; r < 8; ++r)
      out[b0 + 8 * hi + r] = fastsig(p[r] + bf);
  }
}

extern "C" void kernel_launch(void* const* d_in, const int* in_sizes, int n_in,
                              void* d_out, int out_size, void* d_ws,
                              size_t ws_size, hipStream_t stream) {
  (void)in_sizes; (void)n_in; (void)d_ws; (void)ws_size; (void)out_size;
  const float* x    = (const float*)d_in[0];
  const float* Wih1 = (const float*)d_in[1];
  const float* Whh1 = (const float*)d_in[2];
  const float* bih1 = (const float*)d_in[3];
  const float* bhh1 = (const float*)d_in[4];
  const float* Wih2 = (const float*)d_in[5];
  const float* Whh2 = (const float*)d_in[6];
  const float* bih2 = (const float*)d_in[7];
  const float* bhh2 = (const float*)d_in[8];
  const float* Wfc  = (const float*)d_in[9];
  const float* bfc  = (const float*)d_in[10];

  dim3 grid(B_ / 16);   // 128 waves, one 16-row batch tile each
  dim3 block(32);       // one wave32 per workgroup -> barriers are free NOPs
  lstm_fused_kernel<<<grid, block, 0, stream>>>(
      x, Wih1, Whh1, bih1, bhh1, Wih2, Whh2, bih2, bhh2, Wfc, bfc,
      (float*)d_out);
}